// LocalInteraction_32899449487753
// MI455X (gfx1250) — compile-verified
//
#include <hip/hip_runtime.h>

#define FEAT 128
#define NNODES 16000
#define NEDGES 256000

typedef __attribute__((ext_vector_type(2))) float v2f;
typedef __attribute__((ext_vector_type(8))) float v8f;

__device__ __forceinline__ v8f wmma_f32(v2f a, v2f b, v8f c) {
    // D = A(16x4) * B(4x16) + C(16x16), all f32, wave32
    return __builtin_amdgcn_wmma_f32_16x16x4_f32(false, a, false, b, (short)0, c,
                                                 false, false);
}

__device__ __forceinline__ float swish_f(float x, float a, float b) {
    // a * x * sigmoid(b*x); fast rcp (v_rcp_f32) instead of IEEE divide chain
    float e = __expf(-b * x);
    return a * x * __builtin_amdgcn_rcpf(1.0f + e);
}

// ---------------------------------------------------------------------------
// Residual MLP over a 16-node tile. Block = 256 threads = 8 waves; wave w owns
// output-feature tile [16w, 16w+16). Activations staged in LDS.
// ---------------------------------------------------------------------------
__global__ __launch_bounds__(256) void node_mlp_kernel(
    const float* __restrict__ X, float* __restrict__ Y,
    const float* __restrict__ W1, const float* __restrict__ b1,
    const float* __restrict__ W2, const float* __restrict__ b2,
    const float* __restrict__ W3, const float* __restrict__ b3,
    const float* __restrict__ al, const float* __restrict__ be)
{
    __shared__ float xs[16 * FEAT];   // raw input tile (for residual)
    __shared__ float t [16 * FEAT];   // layer-1 input / layer-3 input (reused)
    __shared__ float t2[16 * FEAT];   // layer-2 input

    const int tid   = threadIdx.x;
    const int wave  = tid >> 5;
    const int lane  = tid & 31;
    const int node0 = blockIdx.x * 16;

    // load x tile, apply swish0 -> t
    for (int idx = tid; idx < 16 * FEAT; idx += 256) {
        int m = idx >> 7, f = idx & 127;
        float v = X[(size_t)(node0 + m) * FEAT + f];
        xs[idx] = v;
        t[idx]  = swish_f(v, al[f], be[f]);
    }
    __syncthreads();

    const int n0   = wave * 16;
    const int mrow = (lane & 15) * FEAT;       // A-matrix row base (node)
    const int kb   = (lane >> 4) * 2;          // K sub-offset per ISA layout
    const int ncol = n0 + (lane & 15);         // B/C column (output feature)
    const int rofs = (lane >> 4) * 8;          // C/D row offset

    // ---- layer 1: h1 = swish0(x) @ W1 + b1 ; t2 = swish1(h1)
    {
        v8f acc = {};
        #pragma unroll 8
        for (int ks = 0; ks < 32; ++ks) {
            int k = ks * 4 + kb;
            v2f a = *(const v2f*)&t[mrow + k];
            v2f b;
            b.x = W1[(size_t)k * FEAT + ncol];
            b.y = W1[(size_t)(k + 1) * FEAT + ncol];
            acc = wmma_f32(a, b, acc);
        }
        float bias = b1[ncol], aa = al[FEAT + ncol], bb = be[FEAT + ncol];
        #pragma unroll
        for (int r = 0; r < 8; ++r)
            t2[(r + rofs) * FEAT + ncol] = swish_f(acc[r] + bias, aa, bb);
    }
    __syncthreads();

    // ---- layer 2: h2 = t2 @ W2 + b2 ; t = swish2(x + h2)
    {
        v8f acc = {};
        #pragma unroll 8
        for (int ks = 0; ks < 32; ++ks) {
            int k = ks * 4 + kb;
            v2f a = *(const v2f*)&t2[mrow + k];
            v2f b;
            b.x = W2[(size_t)k * FEAT + ncol];
            b.y = W2[(size_t)(k + 1) * FEAT + ncol];
            acc = wmma_f32(a, b, acc);
        }
        float bias = b2[ncol], aa = al[2 * FEAT + ncol], bb = be[2 * FEAT + ncol];
        #pragma unroll
        for (int r = 0; r < 8; ++r) {
            float rr = xs[(r + rofs) * FEAT + ncol] + acc[r] + bias;
            t[(r + rofs) * FEAT + ncol] = swish_f(rr, aa, bb);
        }
    }
    __syncthreads();

    // ---- layer 3: Y = t @ W3 + b3
    {
        v8f acc = {};
        #pragma unroll 8
        for (int ks = 0; ks < 32; ++ks) {
            int k = ks * 4 + kb;
            v2f a = *(const v2f*)&t[mrow + k];
            v2f b;
            b.x = W3[(size_t)k * FEAT + ncol];
            b.y = W3[(size_t)(k + 1) * FEAT + ncol];
            acc = wmma_f32(a, b, acc);
        }
        float bias = b3[ncol];
        #pragma unroll
        for (int r = 0; r < 8; ++r)
            Y[(size_t)(node0 + r + rofs) * FEAT + ncol] = acc[r] + bias;
    }
}

// ---------------------------------------------------------------------------
// Edge pass: one wave per edge, 4 features per lane. Uses factorization
// mm[f,l] = (sum_k G[f,k]*rad[k]) * Y[l]. Atomic scatter into per-node accums
// (accumulators are L2-resident: 74 MB << 192 MB L2).
// ---------------------------------------------------------------------------
__constant__ float c_binom[16] = {1.f, 15.f, 105.f, 455.f, 1365.f, 3003.f,
                                  5005.f, 6435.f, 6435.f, 5005.f, 3003.f,
                                  1365.f, 455.f, 105.f, 15.f, 1.f};

__global__ __launch_bounds__(256) void edge_kernel(
    const float* __restrict__ xyz, const int* __restrict__ nbrs,
    const float* __restrict__ M1, const float* __restrict__ M2,
    const float* __restrict__ M3,
    const float* __restrict__ Gs, const float* __restrict__ Gp,
    const float* __restrict__ Gd,
    float* __restrict__ inp, float* __restrict__ Qp, float* __restrict__ Qd,
    int n_edges)
{
    const int wid  = (int)((blockIdx.x * 256u + threadIdx.x) >> 5);
    const int lane = threadIdx.x & 31;
    if (wid >= n_edges) return;

    const int i = nbrs[2 * wid + 0];
    const int j = nbrs[2 * wid + 1];

    float rx = xyz[3 * j + 0] - xyz[3 * i + 0];
    float ry = xyz[3 * j + 1] - xyz[3 * i + 1];
    float rz = xyz[3 * j + 2] - xyz[3 * i + 2];
    float d2 = rx * rx + ry * ry + rz * rz + 1e-12f;
    float iv = __builtin_amdgcn_rsqf(d2);   // v_rsq_f32: 1/sqrt(d2)
    float d  = d2 * iv;                     // d = sqrt(d2)
    float ux = rx * iv, uy = ry * iv, uz = rz * iv;
    float dd = d * d;

    // Bernstein radial basis * cutoff
    float rho = __expf(-0.5f * d);
    float omr = 1.0f - rho;
    float pr[16], po[16];
    pr[0] = 1.0f; po[0] = 1.0f;
    #pragma unroll
    for (int k = 1; k < 16; ++k) { pr[k] = pr[k - 1] * rho; po[k] = po[k - 1] * omr; }
    float fcut = (d < 5.0f)
        ? __expf(-dd * __builtin_amdgcn_rcpf(25.0f - dd + 1e-9f))
        : 0.0f;
    float rad[16];
    #pragma unroll
    for (int k = 0; k < 16; ++k) rad[k] = c_binom[k] * pr[k] * po[15 - k] * fcut;

    float Y2[5] = {ux * uy, ux * uz, uy * uz, ux * ux - uy * uy,
                   3.0f * uz * uz - 1.0f};

    const size_t jb = (size_t)j * FEAT;
    const size_t ib = (size_t)i * FEAT;
    const size_t ip = (size_t)i * (3 * FEAT);
    const size_t id = (size_t)i * (5 * FEAT);

    #pragma unroll
    for (int ff = 0; ff < 4; ++ff) {
        int f = lane * 4 + ff;
        const float* gs = Gs + f * 16;
        const float* gp = Gp + f * 16;
        const float* gd = Gd + f * 16;
        float Rs = 0.f, Rp = 0.f, Rd = 0.f;
        #pragma unroll
        for (int k = 0; k < 16; ++k) {
            Rs += gs[k] * rad[k];
            Rp += gp[k] * rad[k];
            Rd += gd[k] * rad[k];
        }
        // s-term straight into inp
        atomicAdd(&inp[ib + f], M1[jb + f] * Rs);
        // p-term into Qp (layout: node, l, g)
        float m2 = M2[jb + f] * Rp;
        atomicAdd(&Qp[ip + 0 * FEAT + f], m2 * ux);
        atomicAdd(&Qp[ip + 1 * FEAT + f], m2 * uy);
        atomicAdd(&Qp[ip + 2 * FEAT + f], m2 * uz);
        // d-term into Qd (layout: node, l, g)
        float m3 = M3[jb + f] * Rd;
        #pragma unroll
        for (int l = 0; l < 5; ++l)
            atomicAdd(&Qd[id + l * FEAT + f], m3 * Y2[l]);
    }
}

// ---------------------------------------------------------------------------
// P/D invariants: per 16-node tile, contract over g=128 with WMMA.
// inp[n,f] += sum_l (P1@Qp)[f,l]*(P2@Qp)[f,l] + sum_l (D1@Qd)[f,l]*(D2@Qd)[f,l]
// ---------------------------------------------------------------------------
__global__ __launch_bounds__(256) void pd_kernel(
    const float* __restrict__ P1, const float* __restrict__ P2,
    const float* __restrict__ D1, const float* __restrict__ D2,
    const float* __restrict__ Qp, const float* __restrict__ Qd,
    float* __restrict__ inp)
{
    const int wave  = threadIdx.x >> 5;
    const int lane  = threadIdx.x & 31;
    const int node0 = blockIdx.x * 16;
    const int f0    = wave * 16;

    const int arow = (f0 + (lane & 15)) * FEAT;             // A row base (feature f)
    const int kb   = (lane >> 4) * 2;
    const size_t nQp = (size_t)(node0 + (lane & 15)) * (3 * FEAT);
    const size_t nQd = (size_t)(node0 + (lane & 15)) * (5 * FEAT);
    const int rofs = (lane >> 4) * 8;

    v8f sum = {};

    // ---- P part (l = 0..2)
    for (int l = 0; l < 3; ++l) {
        v8f a1 = {}, a2 = {};
        #pragma unroll 8
        for (int ks = 0; ks < 32; ++ks) {
            int k = ks * 4 + kb;
            v2f A1 = *(const v2f*)&P1[arow + k];
            v2f A2 = *(const v2f*)&P2[arow + k];
            v2f B  = *(const v2f*)&Qp[nQp + (size_t)l * FEAT + k];
            a1 = wmma_f32(A1, B, a1);
            a2 = wmma_f32(A2, B, a2);
        }
        sum += a1 * a2;
    }

    // ---- D part (l = 0..4)
    for (int l = 0; l < 5; ++l) {
        v8f a1 = {}, a2 = {};
        #pragma unroll 8
        for (int ks = 0; ks < 32; ++ks) {
            int k = ks * 4 + kb;
            v2f A1 = *(const v2f*)&D1[arow + k];
            v2f A2 = *(const v2f*)&D2[arow + k];
            v2f B  = *(const v2f*)&Qd[nQd + (size_t)l * FEAT + k];
            a1 = wmma_f32(A1, B, a1);
            a2 = wmma_f32(A2, B, a2);
        }
        sum += a1 * a2;
    }

    // C/D tile: row = feature f0+r+rofs, col = node (lane&15)
    const size_t nb = (size_t)(node0 + (lane & 15)) * FEAT;
    #pragma unroll
    for (int r = 0; r < 8; ++r)
        inp[nb + f0 + r + rofs] += sum[r];
}

// ---------------------------------------------------------------------------
extern "C" void kernel_launch(void* const* d_in, const int* in_sizes, int n_in,
                              void* d_out, int out_size, void* d_ws, size_t ws_size,
                              hipStream_t stream) {
    (void)in_sizes; (void)n_in; (void)out_size; (void)ws_size;
    const float* xyz     = (const float*)d_in[0];
    const float* x_tilde = (const float*)d_in[1];
    const int*   nbrs    = (const int*)  d_in[2];
    const float* W1      = (const float*)d_in[3];
    const float* b1      = (const float*)d_in[4];
    const float* W2      = (const float*)d_in[5];
    const float* b2      = (const float*)d_in[6];
    const float* W3      = (const float*)d_in[7];
    const float* b3      = (const float*)d_in[8];
    const float* alpha   = (const float*)d_in[9];
    const float* beta    = (const float*)d_in[10];
    const float* G_s     = (const float*)d_in[11];
    const float* G_p     = (const float*)d_in[12];
    const float* G_d     = (const float*)d_in[13];
    const float* P_1     = (const float*)d_in[14];
    const float* P_2     = (const float*)d_in[15];
    const float* D_1     = (const float*)d_in[16];
    const float* D_2     = (const float*)d_in[17];
    float* out = (float*)d_out;

    const size_t NF = (size_t)NNODES * FEAT;
    float* inp = (float*)d_ws;          // N*128  (c_term, then += s/p/d)
    float* M1  = inp + NF;              // N*128  mlp idx1 per node
    float* M2  = M1 + NF;               // N*128  mlp idx2
    float* M3  = M2 + NF;               // N*128  mlp idx3
    float* Qp  = M3 + NF;               // N*3*128
    float* Qd  = Qp + (size_t)NNODES * 3 * FEAT;   // N*5*128

    // zero edge accumulators (Qp and Qd are contiguous)
    hipMemsetAsync(Qp, 0, (size_t)NNODES * 8 * FEAT * sizeof(float), stream);

    const size_t WS = (size_t)FEAT * FEAT;   // per-idx weight stride
    const size_t BS = FEAT;                  // per-idx bias stride
    const size_t AS = 3 * FEAT;              // per-idx alpha/beta stride
    dim3 blk(256);
    dim3 gN(NNODES / 16);

    // idx 0: c_term -> inp ; idx 1..3 -> M1..M3 (node-level hoist of mlp(x_j))
    node_mlp_kernel<<<gN, blk, 0, stream>>>(x_tilde, inp,
        W1 + 0 * WS, b1 + 0 * BS, W2 + 0 * WS, b2 + 0 * BS, W3 + 0 * WS, b3 + 0 * BS,
        alpha + 0 * AS, beta + 0 * AS);
    node_mlp_kernel<<<gN, blk, 0, stream>>>(x_tilde, M1,
        W1 + 1 * WS, b1 + 1 * BS, W2 + 1 * WS, b2 + 1 * BS, W3 + 1 * WS, b3 + 1 * BS,
        alpha + 1 * AS, beta + 1 * AS);
    node_mlp_kernel<<<gN, blk, 0, stream>>>(x_tilde, M2,
        W1 + 2 * WS, b1 + 2 * BS, W2 + 2 * WS, b2 + 2 * BS, W3 + 2 * WS, b3 + 2 * BS,
        alpha + 2 * AS, beta + 2 * AS);
    node_mlp_kernel<<<gN, blk, 0, stream>>>(x_tilde, M3,
        W1 + 3 * WS, b1 + 3 * BS, W2 + 3 * WS, b2 + 3 * BS, W3 + 3 * WS, b3 + 3 * BS,
        alpha + 3 * AS, beta + 3 * AS);

    // edge scatter (s-term into inp, p/d moments into Qp/Qd)
    edge_kernel<<<NEDGES / 8, blk, 0, stream>>>(xyz, nbrs, M1, M2, M3,
                                                G_s, G_p, G_d, inp, Qp, Qd, NEDGES);

    // p/d invariants into inp
    pd_kernel<<<gN, blk, 0, stream>>>(P_1, P_2, D_1, D_2, Qp, Qd, inp);

    // final mlp idx 4: inp -> out
    node_mlp_kernel<<<gN, blk, 0, stream>>>(inp, out,
        W1 + 4 * WS, b1 + 4 * BS, W2 + 4 * WS, b2 + 4 * BS, W3 + 4 * WS, b3 + 4 * BS,
        alpha + 4 * AS, beta + 4 * AS);
}